// OmniContrastiveFeatureLoss_24515673326329
// MI455X (gfx1250) — compile-verified
//
#include <hip/hip_runtime.h>
#include <math.h>

// ---------------------------------------------------------------------------
// OmniContrastiveFeatureLoss for MI455X (gfx1250, wave32)
//   - f32 WMMA 16x16x4 for the align 1x1 conv and both 3x3 projector convs
//   - Tensor Data Mover (tensor_load_to_lds) for LDS tile staging, with
//     TDM pad fields producing bank-padded LDS row strides directly
// ---------------------------------------------------------------------------

typedef float v2f  __attribute__((ext_vector_type(2)));
typedef float v8f  __attribute__((ext_vector_type(8)));
typedef unsigned int u32x4 __attribute__((ext_vector_type(4)));
typedef int          i32x4 __attribute__((ext_vector_type(4)));
typedef int          i32x8 __attribute__((ext_vector_type(8)));

#define TAU   0.07f
#define CS    512
#define CT    2048
#define HH    64
#define WW    64
#define NB    2          // batch
#define PH    16         // pooled H
#define PW    16         // pooled W

// D = A(16x4) * B(4x16) + C, full f32 precision (matches reference dtype).
__device__ __forceinline__ v8f wmma4_f32(v2f a, v2f b, v8f c) {
  return __builtin_amdgcn_wmma_f32_16x16x4_f32(false, a, false, b, (short)0, c,
                                               false, false);
}

// ---------------------------------------------------------------------------
// TDM 2D tile load: tile (tile0 x tile1) f32 elements, global row stride
// stride0 (elements), into LDS at lds_addr.  pad_interval/pad_amount are the
// raw D# field encodings (interval: v -> 2^(v+1) dwords; amount: v -> v+1
// dwords), used to create padded LDS row strides.
// D# packing per CDNA5 ISA 8.3/8.4 (group0: count/lds/global/type,
// group1: data_size, dims, tile dims, stride).  Groups 2/3 zero (2D tile).
// Wave-uniform; call from one wave only (EXEC is ignored by TDM).
// This toolchain exposes the 6-arg builtin (extra i32x8 group before cpol).
// ---------------------------------------------------------------------------
__device__ __forceinline__ void tdm_load_2d(unsigned lds_addr, const float* gptr,
                                            unsigned tile0, unsigned tile1,
                                            unsigned long long stride0,
                                            unsigned pad_en, unsigned pad_interval,
                                            unsigned pad_amount) {
  unsigned long long ga = (unsigned long long)(uintptr_t)gptr;
  u32x4 g0;
  g0.x = 1u;                                         // count=1, user mode
  g0.y = lds_addr;                                   // lds_addr [63:32]
  g0.z = (unsigned)(ga & 0xFFFFFFFFu);               // global_addr lo
  g0.w = (unsigned)((ga >> 32) & 0x01FFFFFFu) | (2u << 30);  // hi | type=2
  i32x8 g1;
  g1[0] = (int)((2u << 16) | (pad_en << 20) | (pad_interval << 22) |
                (pad_amount << 25));                 // data_size=4B + pad cfg
  g1[1] = (int)((tile0 & 0xFFFFu) << 16);            // tensor_dim0 lo16
  g1[2] = (int)((tile0 >> 16) | ((tile1 & 0xFFFFu) << 16)); // d0 hi | d1 lo
  g1[3] = (int)((tile1 >> 16) | (tile0 << 16));      // d1 hi | tile_dim0
  g1[4] = (int)(tile1);                              // tile_dim1 | tile_dim2=0
  g1[5] = (int)(stride0 & 0xFFFFFFFFull);            // tensor_dim0_stride lo
  g1[6] = (int)((stride0 >> 32) & 0xFFFFull);        // stride hi | d1_stride=0
  g1[7] = 0;
  i32x4 gz4 = {0, 0, 0, 0};
  i32x8 gz8 = {0, 0, 0, 0, 0, 0, 0, 0};
  __builtin_amdgcn_tensor_load_to_lds(g0, g1, gz4, gz4, gz8, 0);
}

__device__ __forceinline__ unsigned lds_off(const void* p) {
  return (unsigned)(uintptr_t)p;   // flat LDS address: low 32 bits = offset
}

// ---------------------------------------------------------------------------
// Kernel 1: 1x1 align conv + bias + random mask.  Block tile: 64 co x 64 px
// (one image row).  8 waves: mw=wave&3 (M subtile), nw=wave>>2 (N half).
// Both tiles staged by TDM; pad fields give As stride 17, Bs stride 68.
// ---------------------------------------------------------------------------
__global__ __launch_bounds__(256)
void align_mask_kernel(const float* __restrict__ S, const float* __restrict__ Wa,
                       const float* __restrict__ ba, const float* __restrict__ mrand,
                       float* __restrict__ out) {
  const int y   = blockIdx.x;
  const int co0 = blockIdx.y * 64;
  const int b   = blockIdx.z;

  __shared__ float As[64][17];   // [m][k], TDM: 16 dwords + 1 pad
  __shared__ float Bs[16][68];   // [k][x], TDM: 64 dwords + 4 pad

  const int tid  = threadIdx.x;
  const int lane = tid & 31;
  const int wv   = tid >> 5;
  const int mw   = wv & 3;
  const int nw   = wv >> 2;
  const int hi   = lane >> 4;
  const int l15  = lane & 15;

  v8f acc0 = {}; v8f acc1 = {};
  const float* Sb = S + (size_t)b * CS * HH * WW;

  for (int k0 = 0; k0 < CS; k0 += 16) {
    if (wv == 0) {
      // A: 64 rows (m, stride CS) x 16 floats  -> As[64][17]
      tdm_load_2d(lds_off(&As[0][0]), Wa + (size_t)co0 * CS + k0,
                  16, 64, CS, 1, 3, 0);
      // B: 16 rows (ci, stride 4096) x 64 floats -> Bs[16][68]
      tdm_load_2d(lds_off(&Bs[0][0]), Sb + (size_t)k0 * (HH * WW) + (size_t)y * WW,
                  64, 16, HH * WW, 1, 5, 3);
      __builtin_amdgcn_s_wait_tensorcnt(0);
    }
    __syncthreads();
#pragma unroll
    for (int kk = 0; kk < 16; kk += 4) {
      v2f a, b0, b1;
      a.x  = As[mw * 16 + l15][kk + hi * 2 + 0];
      a.y  = As[mw * 16 + l15][kk + hi * 2 + 1];
      b0.x = Bs[kk + hi * 2 + 0][nw * 32 + l15];
      b0.y = Bs[kk + hi * 2 + 1][nw * 32 + l15];
      b1.x = Bs[kk + hi * 2 + 0][nw * 32 + 16 + l15];
      b1.y = Bs[kk + hi * 2 + 1][nw * 32 + 16 + l15];
      acc0 = wmma4_f32(a, b0, acc0);
      acc1 = wmma4_f32(a, b1, acc1);
    }
    __syncthreads();
  }

  float* Ob = out + (size_t)b * CT * HH * WW;
  const float* Mb = mrand + (size_t)b * HH * WW;
#pragma unroll
  for (int r = 0; r < 8; r++) {
    int row = co0 + mw * 16 + r + hi * 8;   // C/D layout: vgpr r -> M=r / r+8
    float bias = ba[row];
    int c0 = nw * 32 + l15;
    int c1 = c0 + 16;
    float m0 = Mb[y * WW + c0] > 0.25f ? 0.f : 1.f;
    float m1 = Mb[y * WW + c1] > 0.25f ? 0.f : 1.f;
    Ob[(size_t)row * (HH * WW) + y * WW + c0] = (acc0[r] + bias) * m0;
    Ob[(size_t)row * (HH * WW) + y * WW + c1] = (acc1[r] + bias) * m1;
  }
}

// ---------------------------------------------------------------------------
// Kernel 2/3: 3x3 conv as implicit GEMM.  dy outer (OOB rows skipped: their
// B tile is all zero), one haloed B row-tile per K-chunk shared by all 3 dx
// taps (dx becomes an LDS column shift), A staged for all 3 taps at once.
// 24 WMMA per barrier round.
// RELU: fused ReLU epilogue.  DIS: fused sum((out-T)^2) reduce + atomic.
// ---------------------------------------------------------------------------
template <bool RELU, bool DIS>
__global__ __launch_bounds__(256)
void conv3_kernel(const float* __restrict__ X, const float* __restrict__ Wc,
                  const float* __restrict__ bc, const float* __restrict__ T,
                  float* __restrict__ out, float* __restrict__ acc_dis) {
  const int y   = blockIdx.x;
  const int co0 = blockIdx.y * 64;
  const int b   = blockIdx.z;

  __shared__ float As3[3][64][17];  // [dx][m][k]
  __shared__ float Bs[16][68];      // [k][1+x], halo at cols 0 and 65
  __shared__ float red[256];

  const int tid  = threadIdx.x;
  const int lane = tid & 31;
  const int wv   = tid >> 5;
  const int mw   = wv & 3;
  const int nw   = wv >> 2;
  const int hi   = lane >> 4;
  const int l15  = lane & 15;

  // zero halo columns once (x = -1 and x = 64 are always out of range)
  if (tid < 16) { Bs[tid][0] = 0.f; Bs[tid][65] = 0.f; }

  v8f acc0 = {}; v8f acc1 = {};
  const float* Xb = X + (size_t)b * CT * HH * WW;

  for (int dy = 0; dy < 3; ++dy) {
    const int yy = y + dy - 1;
    if ((unsigned)yy >= (unsigned)HH) continue;   // zero B tile -> no work
    for (int k0 = 0; k0 < CT; k0 += 16) {
      if (wv == 0) {
        // B: 16 rows (ci, stride 4096) x 64 floats -> Bs[.][1..64]
        tdm_load_2d(lds_off(&Bs[0][1]),
                    Xb + (size_t)k0 * (HH * WW) + (size_t)yy * WW,
                    64, 16, HH * WW, 1, 5, 3);
      }
      // A for all three dx taps: 12 unconditional loads per thread
      {
        const int m  = tid >> 2;
        const int k4 = (tid & 3) * 4;
        const size_t base = ((size_t)(co0 + m) * CT + (k0 + k4)) * 9 + dy * 3;
#pragma unroll
        for (int dx = 0; dx < 3; ++dx)
#pragma unroll
          for (int kq = 0; kq < 4; ++kq)
            As3[dx][m][k4 + kq] = Wc[base + (size_t)kq * 9 + dx];
      }
      if (wv == 0) __builtin_amdgcn_s_wait_tensorcnt(0);
      __syncthreads();
#pragma unroll
      for (int dx = 0; dx < 3; ++dx) {
#pragma unroll
        for (int kk = 0; kk < 16; kk += 4) {
          v2f a, b0, b1;
          a.x  = As3[dx][mw * 16 + l15][kk + hi * 2 + 0];
          a.y  = As3[dx][mw * 16 + l15][kk + hi * 2 + 1];
          b0.x = Bs[kk + hi * 2 + 0][nw * 32 + l15 + dx];
          b0.y = Bs[kk + hi * 2 + 1][nw * 32 + l15 + dx];
          b1.x = Bs[kk + hi * 2 + 0][nw * 32 + 16 + l15 + dx];
          b1.y = Bs[kk + hi * 2 + 1][nw * 32 + 16 + l15 + dx];
          acc0 = wmma4_f32(a, b0, acc0);
          acc1 = wmma4_f32(a, b1, acc1);
        }
      }
      __syncthreads();
    }
  }

  float* Ob = out + (size_t)b * CT * HH * WW;
  const float* Tb = DIS ? (T + (size_t)b * CT * HH * WW) : nullptr;
  float lsum = 0.f;
#pragma unroll
  for (int r = 0; r < 8; r++) {
    int row = co0 + mw * 16 + r + hi * 8;
    float bias = bc[row];
    int c0 = nw * 32 + l15;
    int c1 = c0 + 16;
    float v0 = acc0[r] + bias;
    float v1 = acc1[r] + bias;
    if (RELU) { v0 = fmaxf(v0, 0.f); v1 = fmaxf(v1, 0.f); }
    size_t o0 = (size_t)row * (HH * WW) + y * WW + c0;
    size_t o1 = (size_t)row * (HH * WW) + y * WW + c1;
    Ob[o0] = v0;
    Ob[o1] = v1;
    if (DIS) {
      float d0 = v0 - Tb[o0];
      float d1 = v1 - Tb[o1];
      lsum += d0 * d0 + d1 * d1;
    }
  }
  if (DIS) {
    red[tid] = lsum;
    __syncthreads();
    for (int st = 128; st > 0; st >>= 1) {
      if (tid < st) red[tid] += red[tid + st];
      __syncthreads();
    }
    if (tid == 0) atomicAdd(acc_dis, red[0]);
  }
}

// ---------------------------------------------------------------------------
// Kernel 4: 4x4 maxpool of new_fea (which=0) and feat_T (which=1).
// ---------------------------------------------------------------------------
__global__ void pool_kernel(const float* __restrict__ NF, const float* __restrict__ T,
                            float* __restrict__ P, float* __restrict__ Q) {
  int idx = blockIdx.x * blockDim.x + threadIdx.x;
  const int total = NB * CT * PH * PW;
  if (idx >= 2 * total) return;
  int which = idx / total;
  int r = idx - which * total;
  int ox = r & 15;
  int oy = (r >> 4) & 15;
  int bc = r >> 8;              // b*CT + c
  const float* src = (which ? T : NF) + (size_t)bc * (HH * WW);
  float m = -INFINITY;
#pragma unroll
  for (int py = 0; py < 4; ++py)
#pragma unroll
    for (int px = 0; px < 4; ++px)
      m = fmaxf(m, src[(oy * 4 + py) * WW + ox * 4 + px]);
  (which ? Q : P)[r] = m;
}

// ---------------------------------------------------------------------------
// Kernel 5: contrastive CE.  One block per patch-batch (32), one thread per
// token (256).  Token i = n2*16 + m (m = channel group).  logsumexp over
// {pos} U {j : j%16 != i%16}, minus pos logit.
// ---------------------------------------------------------------------------
__global__ __launch_bounds__(256)
void ctr_kernel(const float* __restrict__ P, const float* __restrict__ Q,
                float* __restrict__ acc_ce) {
  const int pb = blockIdx.x;
  const int b = pb >> 4, pidx = pb & 15;
  const int hh = pidx >> 2, ww = pidx & 3;
  const int i = threadIdx.x;
  const int mi = i & 15, n2 = i >> 4;
  const int h = hh * 4 + (n2 >> 2), w = ww * 4 + (n2 & 3);
  const float* pc = P + (((size_t)(b * CT + mi * 128) * PH + h) * PW + w);
  // channel stride within pc/tc = PH*PW = 256 floats

  float mx = -INFINITY, s = 0.f, pos = 0.f;
  for (int j = 0; j < 256; ++j) {
    int mj = j & 15;
    if (mj == mi && j != i) continue;         // exclude same-group non-pos
    int n2j = j >> 4;
    int hj = hh * 4 + (n2j >> 2), wj = ww * 4 + (n2j & 3);
    const float* tc = Q + (((size_t)(b * CT + mj * 128) * PH + hj) * PW + wj);
    float d = 0.f;
    for (int c = 0; c < 128; ++c)
      d += fabsf(pc[(size_t)c * 256] - tc[(size_t)c * 256]);
    float lg = -d / TAU;
    if (j == i) pos = lg;
    if (lg > mx) { s = s * expf(mx - lg) + 1.f; mx = lg; }
    else         { s += expf(lg - mx); }
  }
  float li = (mx + logf(s)) - pos;

  __shared__ float red[256];
  red[i] = li;
  __syncthreads();
  for (int st = 128; st > 0; st >>= 1) {
    if (i < st) red[i] += red[i + st];
    __syncthreads();
  }
  if (i == 0) atomicAdd(acc_ce, red[0]);
}

// ---------------------------------------------------------------------------
__global__ void zero_kernel(float* acc) { acc[0] = 0.f; acc[1] = 0.f; }

__global__ void fin_kernel(const float* __restrict__ acc, float* __restrict__ out) {
  out[0] = acc[0] / (float)NB;                         // dis_loss
  out[1] = acc[1] / (float)(PH * PW) / (float)NB;      // ctr_loss
}

// ---------------------------------------------------------------------------
extern "C" void kernel_launch(void* const* d_in, const int* in_sizes, int n_in,
                              void* d_out, int out_size, void* d_ws, size_t ws_size,
                              hipStream_t stream) {
  (void)in_sizes; (void)n_in; (void)out_size; (void)ws_size;

  const float* feat_S  = (const float*)d_in[0];
  const float* feat_T  = (const float*)d_in[1];
  const float* mrand   = (const float*)d_in[2];
  const float* align_w = (const float*)d_in[3];
  const float* align_b = (const float*)d_in[4];
  const float* w1      = (const float*)d_in[5];
  const float* b1      = (const float*)d_in[6];
  const float* w2      = (const float*)d_in[7];
  const float* b2      = (const float*)d_in[8];

  const size_t FEAT_BYTES = (size_t)NB * CT * HH * WW * sizeof(float); // 64 MB
  char* ws = (char*)d_ws;
  float* bufA  = (float*)ws;                    // masked -> later new_fea
  float* bufB  = (float*)(ws + FEAT_BYTES);     // h -> later pooled p,t
  float* poolP = bufB;
  float* poolT = bufB + (size_t)NB * CT * PH * PW;
  float* acc   = (float*)(ws + 2 * FEAT_BYTES); // [0]=dis, [1]=ce

  zero_kernel<<<1, 1, 0, stream>>>(acc);

  dim3 blk(256);
  dim3 g1(HH, CT / 64, NB);   // (row, co-tile, batch)

  align_mask_kernel<<<g1, blk, 0, stream>>>(feat_S, align_w, align_b, mrand, bufA);
  conv3_kernel<true,  false><<<g1, blk, 0, stream>>>(bufA, w1, b1, nullptr, bufB, nullptr);
  conv3_kernel<false, true ><<<g1, blk, 0, stream>>>(bufB, w2, b2, feat_T, bufA, acc);

  int ptotal = 2 * NB * CT * PH * PW;
  pool_kernel<<<(ptotal + 255) / 256, 256, 0, stream>>>(bufA, feat_T, poolP, poolT);

  ctr_kernel<<<32, 256, 0, stream>>>(poolP, poolT, acc + 1);

  fin_kernel<<<1, 1, 0, stream>>>(acc, (float*)d_out);
}